// ExtractSpans_14955076125125
// MI455X (gfx1250) — compile-verified
//
#include <hip/hip_runtime.h>
#include <hip/hip_bf16.h>
#include <stdint.h>

// Problem constants (from reference setup): S=64, N=8192, L=512, K=128
#define N_CAND 8192
#define MAXLEN 512
#define KOUT   128
#define TPB    1024
#define PER_T  (N_CAND / TPB)   // 8 records per thread for the big sort

#define AS1 __attribute__((address_space(1)))
#define AS3 __attribute__((address_space(3)))

typedef int v4i __attribute__((vector_size(16)));   // int4, matches builtin param

__device__ __forceinline__ bool wave_any32(bool p) {
#if __has_builtin(__builtin_amdgcn_ballot_w32)
  return __builtin_amdgcn_ballot_w32(p) != 0u;   // wave32 ballot (gfx1250)
#else
  return __any((int)p) != 0;
#endif
}

// Record layout (64-bit, sorted ascending in LDS):
//   [63:32] = ~orderable(float score)  -> ascending == descending score
//   [31:18] = candidate index (13 bits used)
//   [17:9]  = start (9 bits)
//   [8:0]   = end   (9 bits)
// Tie-break on equal scores: payload ascending => index ascending == stable argsort.

__global__ __launch_bounds__(TPB)
void extract_spans_kernel(const float* __restrict__ scores,
                          const int*   __restrict__ starts,
                          const int*   __restrict__ ends,
                          int*         __restrict__ out)
{
  extern __shared__ char smem[];
  uint64_t* rec   = (uint64_t*)smem;                    // N_CAND * 8  = 65536 B
  int*      s2e   = (int*)(smem + N_CAND * 8);          // MAXLEN * 4  =  2048 B
  int*      e2s   = s2e + MAXLEN;                       // MAXLEN * 4  =  2048 B
  unsigned* selP  = (unsigned*)(e2s + MAXLEN);          // KOUT * 4    =   512 B
  unsigned* outIx = selP + KOUT;                        // KOUT * 4    =   512 B

  const int sent = blockIdx.x;
  const int tid  = threadIdx.x;

  const float* sc = scores + (size_t)sent * N_CAND;
  const int*   st = starts + (size_t)sent * N_CAND;
  const int*   en = ends   + (size_t)sent * N_CAND;

  // gfx1250 cacheline prefetch of our rows (emits global_prefetch_b8)
  __builtin_prefetch(sc + tid, 0, 3);
  __builtin_prefetch(st + tid, 0, 3);
  __builtin_prefetch(en + tid, 0, 3);

  // ---- Phase 1a: build packed 64-bit records in LDS ----
#pragma unroll
  for (int v = 0; v < PER_T; ++v) {
    int i = tid + v * TPB;
    unsigned u = __float_as_uint(sc[i]);
    u = (u & 0x80000000u) ? ~u : (u | 0x80000000u);   // orderable ascending
    unsigned key = ~u;                                 // ascending key == descending score
    unsigned pay = ((unsigned)i << 18) | (((unsigned)st[i]) << 9) | (unsigned)en[i];
    rec[i] = ((uint64_t)key << 32) | (uint64_t)pay;
  }
  // init interval tables: start->latest end (-1), end->earliest start (L)
  if (tid < MAXLEN) { s2e[tid] = -1; e2s[tid] = MAXLEN; }
  __syncthreads();

  // ---- Phase 1b: bitonic sort of 8192 records, ascending ----
  for (unsigned k = 2; k <= N_CAND; k <<= 1) {
    for (unsigned j = k >> 1; j > 0; j >>= 1) {
#pragma unroll
      for (int v = 0; v < PER_T; ++v) {
        int t   = tid + v * TPB;
        int ixj = t ^ (int)j;
        if (ixj > t) {
          uint64_t a = rec[t], b = rec[ixj];
          bool up = (((unsigned)t & k) == 0);
          if (up ? (a > b) : (a < b)) { rec[t] = b; rec[ixj] = a; }
        }
      }
      __syncthreads();
    }
  }

  // ---- Phase 2: greedy non-crossing selection on wave 0 only ----
  if (tid < 32) {
    const int lane = tid;
    int n = 0;
    for (int i = 0; i < N_CAND && n < KOUT; ++i) {
      unsigned p  = (unsigned)rec[i];          // low 32 bits = payload
      int end_i   = (int)(p & 511u);
      int start_i = (int)((p >> 9) & 511u);
      // crossing iff exists j in (st,en] with s2e[j] > en, or j in [st,en) with e2s[j] < st
      bool cross = false;
      for (int base = start_i; base <= end_i; base += 32) {
        int j = base + lane;
        if (j <= end_i) {
          bool c1 = (j > start_i) && (s2e[j] > end_i);
          bool c2 = (j < end_i)   && (e2s[j] < start_i);
          cross = cross || c1 || c2;
        }
      }
      if (!wave_any32(cross)) {
        if (lane == 0) {
          int v0 = s2e[start_i]; s2e[start_i] = (end_i   > v0) ? end_i   : v0;
          int v1 = e2s[end_i];   e2s[end_i]   = (start_i < v1) ? start_i : v1;
          selP[n] = p;
        }
        __builtin_amdgcn_wave_barrier();   // order lane-0 LDS writes vs next reads
        n++;
      }
    }
    // pad unfilled slots with first selected payload (n >= 1 whenever N >= 1)
    __builtin_amdgcn_wave_barrier();
    if (n > 0) {
      unsigned p0 = selP[0];
      for (int s2 = n + lane; s2 < KOUT; s2 += 32) selP[s2] = p0;
    } else {
      for (int s2 = lane; s2 < KOUT; s2 += 32) selP[s2] = 0u;
    }
  }
  __syncthreads();

  // ---- Phase 3: stable sort K selections by (start*512+end, slot) ----
  if (tid < KOUT) {
    unsigned p    = selP[tid];
    unsigned fkey = ((p & 0x3FFFFu) << 7) | (unsigned)tid;  // span key + slot tiebreak (stable)
    rec[tid] = ((uint64_t)fkey << 32) | (uint64_t)p;
  }
  __syncthreads();
  for (unsigned k = 2; k <= KOUT; k <<= 1) {
    for (unsigned j = k >> 1; j > 0; j >>= 1) {
      if (tid < KOUT) {
        int ixj = tid ^ (int)j;
        if (ixj > tid) {
          uint64_t a = rec[tid], b = rec[ixj];
          bool up = (((unsigned)tid & k) == 0);
          if (up ? (a > b) : (a < b)) { rec[tid] = b; rec[ixj] = a; }
        }
      }
      __syncthreads();
    }
  }
  if (tid < KOUT) outIx[tid] = ((unsigned)rec[tid]) >> 18;  // extract candidate index
  __syncthreads();

  // ---- Output: 128 int32 per sentence ----
#if defined(__gfx1250__) && __has_builtin(__builtin_amdgcn_global_store_async_from_lds_b128)
  // Async LDS->global store: wave 0, one b128 per lane covers all 128 ints.
  if (tid < 32) {
    unsigned* g = (unsigned*)out + (size_t)sent * KOUT + tid * 4;
    __builtin_amdgcn_global_store_async_from_lds_b128(
        (AS1 v4i*)g, (AS3 v4i*)(outIx + tid * 4), 0, 0);
  }
  #if __has_builtin(__builtin_amdgcn_s_wait_asynccnt)
    __builtin_amdgcn_s_wait_asynccnt(0);
  #else
    asm volatile("s_wait_asynccnt 0" ::: "memory");
  #endif
#else
  if (tid < KOUT) out[(size_t)sent * KOUT + tid] = (int)outIx[tid];
#endif
}

extern "C" void kernel_launch(void* const* d_in, const int* in_sizes, int n_in,
                              void* d_out, int out_size, void* d_ws, size_t ws_size,
                              hipStream_t stream) {
  (void)n_in; (void)d_ws; (void)ws_size; (void)out_size;
  const float* scores = (const float*)d_in[0];
  const int*   starts = (const int*)d_in[1];
  const int*   ends   = (const int*)d_in[2];
  int*         out    = (int*)d_out;   // reference output dtype is int32

  int S = in_sizes[0] / N_CAND;        // 64 sentences
  size_t smemBytes = (size_t)N_CAND * 8           // sort records
                   + (size_t)MAXLEN * 4 * 2       // s2e / e2s
                   + (size_t)KOUT * 4 * 2;        // selP / outIx  (= 70656 B, < 320 KB WGP LDS)

  extract_spans_kernel<<<S, TPB, smemBytes, stream>>>(scores, starts, ends, out);
}